// MaskedControlUnit_21577915695732
// MI455X (gfx1250) — compile-verified
//
#include <hip/hip_runtime.h>
#include <math.h>

#define DIMD 768
#define NBATCH 32
#define SLEN 2048
#define NSEG 8
#define SCALE_F 0.03608439182435161f   /* 1/sqrt(768) */

typedef float v2f __attribute__((ext_vector_type(2)));
typedef float v8f __attribute__((ext_vector_type(8)));

// ---------------- reduction helpers (wave32) ----------------
__device__ __forceinline__ float warp_sum(float v) {
#pragma unroll
  for (int off = 16; off; off >>= 1) v += __shfl_down(v, off, 32);
  return v;
}
__device__ __forceinline__ float warp_max(float v) {
#pragma unroll
  for (int off = 16; off; off >>= 1) v = fmaxf(v, __shfl_down(v, off, 32));
  return v;
}
__device__ float block_sum(float v, float* sred) {
  int lane = threadIdx.x & 31, wid = threadIdx.x >> 5;
  v = warp_sum(v);
  if (lane == 0) sred[wid] = v;
  __syncthreads();
  float r = 0.f;
  if (threadIdx.x < (blockDim.x >> 5)) r = sred[threadIdx.x];
  if (wid == 0) r = warp_sum(r);
  if (threadIdx.x == 0) sred[0] = r;
  __syncthreads();
  r = sred[0];
  __syncthreads();
  return r;
}
__device__ float block_max(float v, float* sred) {
  int lane = threadIdx.x & 31, wid = threadIdx.x >> 5;
  v = warp_max(v);
  if (lane == 0) sred[wid] = v;
  __syncthreads();
  float r = -3.0e38f;
  if (threadIdx.x < (blockDim.x >> 5)) r = sred[threadIdx.x];
  if (wid == 0) r = warp_max(r);
  if (threadIdx.x == 0) sred[0] = r;
  __syncthreads();
  r = sred[0];
  __syncthreads();
  return r;
}

// ---------------- WMMA fp32 GEMM: C(MxN) = A(MxK) @ B(KxN) ----------------
// One wave (32 threads) per 16x16 output tile, K marched in steps of 4 via
// V_WMMA_F32_16X16X4_F32.  A 16x4 layout: lane l -> row m=l&15, kpair=(l>>4)*2.
// B 4x16 layout mirrors it along N.  C/D: 8 VGPRs, row = r + 8*(l>>4).
__global__ void gemm_f32_wmma(const float* __restrict__ A,
                              const float* __restrict__ B,
                              int M, int N, int K,
                              const int* __restrict__ stepSel, int stepStride,
                              float* __restrict__ C) {
  (void)M;
  const int lane  = threadIdx.x;        // 0..31, block = 1 wave, EXEC all ones
  const int ntile = blockIdx.x;
  const int mtile = blockIdx.y;
  if (stepSel) B += (size_t)(*stepSel) * (size_t)stepStride;

  const int mrow = mtile * 16 + (lane & 15);
  const int ncol = ntile * 16 + (lane & 15);
  const int kb   = (lane >> 4) * 2;     // 0 or 2

  v8f acc = {};
  for (int k = 0; k < K; k += 4) {
    v2f a, b;
    a.x = A[(size_t)mrow * K + (k + kb)];
    a.y = A[(size_t)mrow * K + (k + kb + 1)];
    b.x = B[(size_t)(k + kb) * N + ncol];
    b.y = B[(size_t)(k + kb + 1) * N + ncol];
    acc = __builtin_amdgcn_wmma_f32_16x16x4_f32(false, a, false, b,
                                                (short)0, acc, false, false);
  }
  const int mbase = mtile * 16 + ((lane >> 4) << 3);
#pragma unroll
  for (int r = 0; r < 8; ++r)
    C[(size_t)(mbase + r) * N + ncol] = acc[r];
}

// ---------------- LayerNorm + ReLU over 768 cols, one block per row ----------
// Optional per-column bias (with device-side step select), dual gamma/beta with
// row split (for the fused ctrl1/ctrl2 pass), scattered output, and optional
// v = y * W_attn side-product (for the cq pass).
__global__ void ln_relu_kernel(const float* __restrict__ X,
                               const float* __restrict__ bias,
                               const int* __restrict__ stepSel, int biasStride,
                               const float* __restrict__ g1, const float* __restrict__ be1,
                               const float* __restrict__ g2, const float* __restrict__ be2,
                               int split,
                               float* __restrict__ Y, int ystride, int yoff2,
                               const float* __restrict__ wattn,
                               float* __restrict__ vout) {
  __shared__ float sred[32];
  const int row = blockIdx.x;
  const float* g  = (row < split) ? g1 : g2;
  const float* be = (row < split) ? be1 : be2;
  const float* bp = bias ? (bias + (stepSel ? (size_t)(*stepSel) * biasStride : 0)) : nullptr;

  float vals[3];
  float s = 0.f;
#pragma unroll
  for (int i = 0; i < 3; ++i) {
    int c = threadIdx.x + 256 * i;
    float x = X[(size_t)row * DIMD + c];
    if (bp) x += bp[c];
    vals[i] = x;
    s += x;
  }
  const float mu = block_sum(s, sred) * (1.0f / DIMD);
  float s2 = 0.f;
#pragma unroll
  for (int i = 0; i < 3; ++i) { float d = vals[i] - mu; s2 += d * d; }
  const float var = block_sum(s2, sred) * (1.0f / DIMD);
  const float inv = rsqrtf(var + 1e-5f);

  const int orow  = (row < split) ? row : (row - split);
  const size_t ob = (size_t)orow * ystride + ((row < split) ? 0 : yoff2);
#pragma unroll
  for (int i = 0; i < 3; ++i) {
    int c = threadIdx.x + 256 * i;
    float y = (vals[i] - mu) * inv * g[c] + be[c];
    y = fmaxf(y, 0.f);
    Y[ob + c] = y;
    if (vout) vout[(size_t)row * DIMD + c] = y * wattn[c];
  }
}

// ---------------- pack [control | pa] -> A2 (32 x 2304) ----------------
__global__ void pack_cq_kernel(const float* __restrict__ control,
                               const float* __restrict__ pa,
                               float* __restrict__ out) {
  int i = blockIdx.x * blockDim.x + threadIdx.x;
  if (i >= NBATCH * 3 * DIMD) return;
  int r = i / (3 * DIMD), c = i % (3 * DIMD);
  out[i] = (c < 2 * DIMD) ? control[(size_t)r * 2 * DIMD + c]
                          : pa[(size_t)r * DIMD + (c - 2 * DIMD)];
}

// ---------------- pass 1 over context: w[b,s] -----------------------------
// One wave per (b,s) row; 6 float4 loads per lane (768 = 32*4*6), shuffle sum.
__global__ void attn_score_kernel(const float* __restrict__ context,
                                  const float* __restrict__ v,
                                  const float* __restrict__ b_attn,
                                  float* __restrict__ w) {
  const int wrow = blockIdx.x * (blockDim.x >> 5) + (threadIdx.x >> 5);
  const int lane = threadIdx.x & 31;
  const int b = wrow >> 11;                      // /2048
  const float4* crow = (const float4*)(context + (size_t)wrow * DIMD);
  const float4* vrow = (const float4*)(v + (size_t)b * DIMD);
  float acc = 0.f;
#pragma unroll
  for (int i = 0; i < 6; ++i) {
    float4 c = crow[lane + 32 * i];
    float4 vv = vrow[lane + 32 * i];
    acc += c.x * vv.x + c.y * vv.y + c.z * vv.z + c.w * vv.w;
  }
  acc = warp_sum(acc);
  if (lane == 0) w[wrow] = (acc + b_attn[0]) * SCALE_F;
}

// ---------------- 3-way masked softmax per batch --------------------------
__global__ void softmax_kernel(const float* __restrict__ w,
                               const float* __restrict__ qm1,
                               const float* __restrict__ qm2,
                               float* __restrict__ attn_out) {
  __shared__ float sred[32];
  const int b = blockIdx.x;
  float l0[8], l1[8], l2[8];
  float m0 = -3.0e38f, m1 = -3.0e38f, m2 = -3.0e38f;
#pragma unroll
  for (int i = 0; i < 8; ++i) {
    int s = threadIdx.x + 256 * i;
    float ww = w[(size_t)b * SLEN + s];
    float q1 = qm1[(size_t)b * SLEN + s];
    float q2 = qm2[(size_t)b * SLEN + s];
    l0[i] = ((q1 + q2) != 0.f) ? ww : -1.0e9f;
    l1[i] = (q1 != 0.f) ? ww : -1.0e9f;
    l2[i] = (q2 != 0.f) ? ww : -1.0e9f;
    m0 = fmaxf(m0, l0[i]); m1 = fmaxf(m1, l1[i]); m2 = fmaxf(m2, l2[i]);
  }
  m0 = block_max(m0, sred); m1 = block_max(m1, sred); m2 = block_max(m2, sred);
  float s0 = 0.f, s1 = 0.f, s2 = 0.f;
#pragma unroll
  for (int i = 0; i < 8; ++i) {
    l0[i] = expf(l0[i] - m0); s0 += l0[i];
    l1[i] = expf(l1[i] - m1); s1 += l1[i];
    l2[i] = expf(l2[i] - m2); s2 += l2[i];
  }
  s0 = block_sum(s0, sred); s1 = block_sum(s1, sred); s2 = block_sum(s2, sred);
  const float r0 = 1.f / s0, r1 = 1.f / s1, r2 = 1.f / s2;
#pragma unroll
  for (int i = 0; i < 8; ++i) {
    int s = threadIdx.x + 256 * i;
    attn_out[((size_t)b * 3 + 0) * SLEN + s] = l0[i] * r0;
    attn_out[((size_t)b * 3 + 1) * SLEN + s] = l1[i] * r1;
    attn_out[((size_t)b * 3 + 2) * SLEN + s] = l2[i] * r2;
  }
}

// ---------------- pass 2 over context: segmented weighted sums ------------
// grid = seg(8) x kc(3) x b(32); each block: 256 k-cols x 256 s-rows.
__global__ void wsum_partial_kernel(const float* __restrict__ context,
                                    const float* __restrict__ attn,
                                    const float* __restrict__ qm1,
                                    const float* __restrict__ qm2,
                                    float* __restrict__ partial) {
  __shared__ float sa0[256], sa1[256], sa2[256];
  const int t   = threadIdx.x;
  const int seg = blockIdx.x & (NSEG - 1);
  const int kc  = (blockIdx.x >> 3) % 3;
  const int b   = blockIdx.x / (3 * NSEG);
  const int k   = kc * 256 + t;
  const int s0  = seg * 256;

  const int sl = s0 + t;
  sa0[t] = attn[((size_t)b * 3 + 0) * SLEN + sl];
  sa1[t] = attn[((size_t)b * 3 + 1) * SLEN + sl] * qm1[(size_t)b * SLEN + sl];
  sa2[t] = attn[((size_t)b * 3 + 2) * SLEN + sl] * qm2[(size_t)b * SLEN + sl];
  __syncthreads();

  const float* ctx = context + (size_t)b * SLEN * DIMD;
  float a0 = 0.f, a1 = 0.f, a2 = 0.f;
#pragma unroll 4
  for (int i = 0; i < 256; ++i) {
    float c = ctx[(size_t)(s0 + i) * DIMD + k];
    a0 += sa0[i] * c;
    a1 += sa1[i] * c;
    a2 += sa2[i] * c;
  }
  const size_t base = ((size_t)b * NSEG + seg) * 3;
  partial[(base + 0) * DIMD + k] = a0;
  partial[(base + 1) * DIMD + k] = a1;
  partial[(base + 2) * DIMD + k] = a2;
}

// reduce segments + L2-normalize; grid = 32*3 blocks (b-major), out [j][b][768]
__global__ void wsum_finalize_kernel(const float* __restrict__ partial,
                                     float* __restrict__ mcn) {
  __shared__ float sred[32];
  const int b = blockIdx.x / 3;
  const int j = blockIdx.x % 3;
  float vals[3];
  float ss = 0.f;
#pragma unroll
  for (int i = 0; i < 3; ++i) {
    int k = threadIdx.x + 256 * i;
    float v = 0.f;
#pragma unroll
    for (int seg = 0; seg < NSEG; ++seg)
      v += partial[(((size_t)b * NSEG + seg) * 3 + j) * DIMD + k];
    vals[i] = v;
    ss += v * v;
  }
  ss = block_sum(ss, sred);
  const float inv = 1.0f / fmaxf(sqrtf(ss), 1e-12f);
#pragma unroll
  for (int i = 0; i < 3; ++i) {
    int k = threadIdx.x + 256 * i;
    mcn[((size_t)j * NBATCH + b) * DIMD + k] = vals[i] * inv;
  }
}

// pack stacked fuse input: rows 0-31 = [mc1|ncf], rows 32-63 = [mc2|ncf]
__global__ void pack_fuse_kernel(const float* __restrict__ mcn,
                                 float* __restrict__ out) {
  int i = blockIdx.x * blockDim.x + threadIdx.x;
  if (i >= 64 * 2 * DIMD) return;
  int r = i / (2 * DIMD), c = i % (2 * DIMD);
  const float* ncf = mcn;
  const float* mc1 = mcn + (size_t)NBATCH * DIMD;
  const float* mc2 = mcn + (size_t)2 * NBATCH * DIMD;
  int bb = (r < NBATCH) ? r : r - NBATCH;
  float v;
  if (c < DIMD) v = ((r < NBATCH) ? mc1 : mc2)[(size_t)bb * DIMD + c];
  else          v = ncf[(size_t)bb * DIMD + (c - DIMD)];
  out[i] = v;
}

// ---------------- launch --------------------------------------------------
extern "C" void kernel_launch(void* const* d_in, const int* in_sizes, int n_in,
                              void* d_out, int out_size, void* d_ws, size_t ws_size,
                              hipStream_t stream) {
  (void)in_sizes; (void)n_in; (void)out_size; (void)ws_size;
  const int*   step    = (const int*)  d_in[0];
  const float* context = (const float*)d_in[1];
  const float* question= (const float*)d_in[2];
  const float* control = (const float*)d_in[3];
  const float* qm1     = (const float*)d_in[4];
  const float* qm2     = (const float*)d_in[5];
  const float* W_pos   = (const float*)d_in[8];
  const float* b_pos   = (const float*)d_in[9];
  const float* ln1_g   = (const float*)d_in[10];
  const float* ln1_b   = (const float*)d_in[11];
  const float* W_cq    = (const float*)d_in[12];
  const float* b_cq    = (const float*)d_in[13];
  const float* ln2_g   = (const float*)d_in[14];
  const float* ln2_b   = (const float*)d_in[15];
  const float* W_attn  = (const float*)d_in[16];
  const float* b_attn  = (const float*)d_in[17];
  const float* W_fuse  = (const float*)d_in[18];
  const float* b_fuse  = (const float*)d_in[19];
  const float* ln3_g   = (const float*)d_in[20];
  const float* ln3_b   = (const float*)d_in[21];
  const float* ln4_g   = (const float*)d_in[22];
  const float* ln4_b   = (const float*)d_in[23];

  float* out      = (float*)d_out;
  float* attn_out = out + NBATCH * 2 * DIMD;   // 49152

  float* ws     = (float*)d_ws;
  float* pa_pre = ws;                 // 24576
  float* pa     = ws + 24576;         // 24576
  float* A2     = ws + 49152;         // 73728
  float* cq_pre = ws + 122880;        // 24576
  float* cq     = ws + 147456;        // 24576
  float* vattn  = ws + 172032;        // 24576
  float* wsc    = ws + 196608;        // 65536
  float* part   = ws + 262144;        // 589824
  float* mcn    = ws + 851968;        // 73728
  float* A3     = ws + 925696;        // 98304
  float* fpre   = ws + 1024000;       // 49152

  // 1) pa_pre = question @ W_pos[step]
  gemm_f32_wmma<<<dim3(DIMD / 16, NBATCH / 16), 32, 0, stream>>>(
      question, W_pos, NBATCH, DIMD, DIMD, step, DIMD * DIMD, pa_pre);
  // 2) pa = LN_ReLU(pa_pre + b_pos[step])
  ln_relu_kernel<<<NBATCH, 256, 0, stream>>>(
      pa_pre, b_pos, step, DIMD, ln1_g, ln1_b, ln1_g, ln1_b, 1 << 30,
      pa, DIMD, 0, nullptr, nullptr);
  // 3) A2 = [control | pa]
  pack_cq_kernel<<<(NBATCH * 3 * DIMD + 255) / 256, 256, 0, stream>>>(control, pa, A2);
  // 4) cq_pre = A2 @ W_cq
  gemm_f32_wmma<<<dim3(DIMD / 16, NBATCH / 16), 32, 0, stream>>>(
      A2, W_cq, NBATCH, DIMD, 3 * DIMD, nullptr, 0, cq_pre);
  // 5) cq = LN_ReLU(cq_pre + b_cq); vattn = cq * W_attn
  ln_relu_kernel<<<NBATCH, 256, 0, stream>>>(
      cq_pre, b_cq, nullptr, 0, ln2_g, ln2_b, ln2_g, ln2_b, 1 << 30,
      cq, DIMD, 0, W_attn, vattn);
  // 6) pass 1: w[b,s]
  attn_score_kernel<<<(NBATCH * SLEN) / 8, 256, 0, stream>>>(context, vattn, b_attn, wsc);
  // 7) softmax -> attentions region of d_out
  softmax_kernel<<<NBATCH, 256, 0, stream>>>(wsc, qm1, qm2, attn_out);
  // 8) pass 2: segmented weighted sums
  wsum_partial_kernel<<<NBATCH * 3 * NSEG, 256, 0, stream>>>(context, attn_out, qm1, qm2, part);
  // 9) reduce + l2norm -> mcn = [ncf | mc1 | mc2]
  wsum_finalize_kernel<<<NBATCH * 3, 256, 0, stream>>>(part, mcn);
  // 10) A3 = stacked fuse input (64 x 1536)
  pack_fuse_kernel<<<(64 * 2 * DIMD + 255) / 256, 256, 0, stream>>>(mcn, A3);
  // 11) fpre = A3 @ W_fuse
  gemm_f32_wmma<<<dim3(DIMD / 16, 64 / 16), 32, 0, stream>>>(
      A3, W_fuse, 64, DIMD, 2 * DIMD, nullptr, 0, fpre);
  // 12) LN_ReLU with ln3 (rows<32) / ln4 (rows>=32), scatter into next_control
  ln_relu_kernel<<<64, 256, 0, stream>>>(
      fpre, b_fuse, nullptr, 0, ln3_g, ln3_b, ln4_g, ln4_b, NBATCH,
      out, 2 * DIMD, DIMD, nullptr, nullptr);
}